// MyModel_78176994722029
// MI455X (gfx1250) — compile-verified
//
#include <hip/hip_runtime.h>

// ---------------------------------------------------------------------------
// MI455X (gfx1250) implementation of the PHT soft-structure-element network.
// Dominant stage (~70-118M exp evals after prefix-mask skipping) is
// TRANS-bound -> offload the exponent argument (rank-5 fp32 GEMM over
// features [s^2, s, 1, y^2, y]) to the matrix pipe with
// V_WMMA_F32_16X16X4_F32; log2(e) is folded into the WMMA B-fragment so the
// loop runs bare v_exp_f32. Wave-uniform values are pinned to SGPRs with
// readfirstlane so the tile loop is scalar-controlled (EXEC stays all-ones
// across WMMA, as the ISA requires). Downstream GEMMs (l1 K=75, fc1 K=200,
// fc2 K=100) also run on fp32 WMMA.
// ---------------------------------------------------------------------------

typedef __attribute__((ext_vector_type(2))) float v2f;
typedef __attribute__((ext_vector_type(8))) float v8f;

static constexpr int V   = 8;
static constexpr int B   = 128;
static constexpr int P   = 512;
static constexpr int NE  = 75;
static constexpr int NEP = 80;     // NE padded to tile multiple
static constexpr float NU  = 0.1f;
static constexpr float EPS = 1e-5f;
static constexpr float INV_SQRT2  = 0.70710678118654752f;
static constexpr float NEG_LOG_NU = 2.302585092994046f;   // -log(0.1)
static constexpr float LOG2E      = 1.4426950408889634f;  // folded into coeffs

__device__ __forceinline__ v8f wmma4(v2f a, v2f b, v8f c) {
  // D = A(16x4,f32) * B(4x16,f32) + C(16x16,f32)
  return __builtin_amdgcn_wmma_f32_16x16x4_f32(
      /*neg_a=*/false, a, /*neg_b=*/false, b,
      /*c_mod=*/(short)0, c, /*reuse_a=*/false, /*reuse_b=*/false);
}

// ---- Stage 1: diagram transform + feature expansion -----------------------
// tp = rotate45(points); y-axis log-thresholded. Store {s^2, s, y^2, y}.
__global__ void feat_kernel(const float* __restrict__ pts,
                            float4* __restrict__ feat, int n) {
  int idx = blockIdx.x * blockDim.x + threadIdx.x;
  if (idx >= n) return;
  float x = pts[2 * idx + 0], y = pts[2 * idx + 1];
  float s = (x + y) * INV_SQRT2;
  float t = (y - x) * INV_SQRT2;
  float ty = (t <= NU) ? (__logf(fmaxf(t, 1e-12f)) + NEG_LOG_NU + NU) : t;
  feat[idx] = make_float4(s * s, s, ty * ty, ty);
}

// ---- Stage 2: WMMA exponent-arg GEMM + exp2 + masked sum over points ------
// One wave per (i, j, b, ne-tile). sl[i][j][b][ne] = sum_p exp(arg) * mask.
__global__ void resp_kernel(const float4* __restrict__ feat,
                            const int* __restrict__ lengths,
                            const float* __restrict__ centers,
                            const float* __restrict__ sharp,
                            float* __restrict__ sl) {
  // Pin the wave id to an SGPR: everything derived from it is then provably
  // wave-uniform -> scalar loop control + saddr-form addressing.
  const int wid = __builtin_amdgcn_readfirstlane(
      blockIdx.x * (blockDim.x >> 5) + (threadIdx.x >> 5));
  const int lane = threadIdx.x & 31;
  const int half = lane >> 4;      // lane half selects K pairs / M+8 rows
  const int l16  = lane & 15;
  int t = wid;
  const int nt = t % 5;   t /= 5;
  const int b  = t % B;   t /= B;
  const int j  = t % 3;   t /= 3;
  const int i  = t;
  if (i >= V) return;                       // wave-uniform
  const int isrc = (i + j + V - 1) % V;     // j=0:(i-1)%V  j=1:i  j=2:(i+1)%V
  const int len  = __builtin_amdgcn_readfirstlane(lengths[isrc * B + b]);
  const int ne   = nt * 16 + l16;

  // Coefficients of arg over features {s^2, s, 1, 0 | y^2, y, 0, 0},
  // pre-scaled by log2(e) so the loop uses raw v_exp_f32 (exp2).
  float k0 = 0.f, k1 = 0.f, k2 = 0.f, k4 = 0.f, k5 = 0.f;
  if (ne < NE) {
    const float c0 = centers[(i * NE + ne) * 2 + 0];
    const float c1 = centers[(i * NE + ne) * 2 + 1];
    const float s0 = sharp[(i * NE + ne) * 2 + 0] * LOG2E;
    const float s1 = sharp[(i * NE + ne) * 2 + 1] * LOG2E;
    k0 = -s0; k1 = 2.f * s0 * c0;
    k4 = -s1; k5 = 2.f * s1 * c1;
    k2 = -(s0 * c0 * c0 + s1 * c1 * c1);
  }
  // B fragments (4x16): VGPR0 holds K even rows, VGPR1 K odd rows per half
  v2f bf0, bf1;
  bf0.x = half ? k2 : k0;  bf0.y = half ? 0.f : k1;
  bf1.x = half ? 0.f : k4; bf1.y = half ? 0.f : k5;

  const float4* fp = feat + (size_t)(isrc * B + b) * P;

  float acc[8];
#pragma unroll
  for (int r = 0; r < 8; ++r) acc[r] = 0.f;

  const int nfull = len >> 4;   // full 16-point tiles (prefix mask -> skip!)
  const int rem   = len & 15;
  for (int tt = 0; tt < nfull; ++tt) {       // scalar-controlled loop
    float4 f = fp[tt * 16 + l16];
    __builtin_prefetch(fp + tt * 16 + l16 + 64, 0, 1);  // global_prefetch_b8
    v2f a0, a1;   // A 16x4: lane m=l16, VGPR0=K{0,2}, VGPR1=K{1,3}
    a0.x = half ? 1.f : f.x;  a0.y = half ? 0.f : f.y;
    a1.x = half ? 0.f : f.z;  a1.y = half ? 0.f : f.w;
    v8f c = {};
    c = wmma4(a0, bf0, c);
    c = wmma4(a1, bf1, c);
#pragma unroll
    for (int r = 0; r < 8; ++r) acc[r] += __builtin_amdgcn_exp2f(c[r]);
  }
  if (rem) {  // uniform branch; per-value predication via select (EXEC full)
    float4 f = fp[nfull * 16 + l16];
    v2f a0, a1;
    a0.x = half ? 1.f : f.x;  a0.y = half ? 0.f : f.y;
    a1.x = half ? 0.f : f.z;  a1.y = half ? 0.f : f.w;
    v8f c = {};
    c = wmma4(a0, bf0, c);
    c = wmma4(a1, bf1, c);
#pragma unroll
    for (int r = 0; r < 8; ++r)
      acc[r] += ((r + 8 * half) < rem) ? __builtin_amdgcn_exp2f(c[r]) : 0.f;
  }
  float tot = 0.f;
#pragma unroll
  for (int r = 0; r < 8; ++r) tot += acc[r];   // sum rows M=r(+8*half)
  tot += __shfl_xor(tot, 16, 32);              // fold the two lane halves
  if (lane < 16 && ne < NE)
    sl[(size_t)((i * 3 + j) * B + b) * NEP + ne] = tot;
}

// ---- Stage 3: conv1(3->16) -> conv2(16->4) -> max over 4 ------------------
__global__ void conv_kernel(const float* __restrict__ sl,
                            const float* __restrict__ w1,
                            const float* __restrict__ w2,
                            float* __restrict__ hbuf) {
  int idx = blockIdx.x * blockDim.x + threadIdx.x;
  if (idx >= V * B * NEP) return;
  const int ne = idx % NEP;
  const int b  = (idx / NEP) % B;
  const int i  = idx / (NEP * B);
  if (ne >= NE) { hbuf[idx] = 0.f; return; }   // zero padding for l1 K-loop
  const float x0 = sl[(size_t)((i * 3 + 0) * B + b) * NEP + ne];
  const float x1 = sl[(size_t)((i * 3 + 1) * B + b) * NEP + ne];
  const float x2 = sl[(size_t)((i * 3 + 2) * B + b) * NEP + ne];
  float tt[16];
#pragma unroll
  for (int f = 0; f < 16; ++f) {
    const float* w = w1 + (i * 16 + f) * 3;
    tt[f] = w[0] * x0 + w[1] * x1 + w[2] * x2;
  }
  float m = -3.4e38f;
#pragma unroll
  for (int g = 0; g < 4; ++g) {
    const float* w = w2 + (i * 4 + g) * 16;
    float u = 0.f;
#pragma unroll
    for (int f = 0; f < 16; ++f) u += w[f] * tt[f];
    m = fmaxf(m, u);
  }
  hbuf[idx] = m;   // [i][b][NEP]
}

// ---- Stage 4: l1 (a = h @ W^T + b), M=128 N=25 K=75 via WMMA --------------
__global__ void l1_kernel(const float* __restrict__ hbuf,
                          const float* __restrict__ l1w,
                          const float* __restrict__ l1b,
                          float* __restrict__ abuf) {
  const int wid = __builtin_amdgcn_readfirstlane(
      blockIdx.x * (blockDim.x >> 5) + (threadIdx.x >> 5));
  const int lane = threadIdx.x & 31;
  const int half = lane >> 4, l16 = lane & 15;
  const int nt2 = wid & 1;
  const int mt  = (wid >> 1) & 7;
  const int i   = wid >> 4;
  if (i >= V) return;
  const int o = nt2 * 16 + l16;
  const float* hp = hbuf + (size_t)(i * B + mt * 16 + l16) * NEP;
  v8f c = {};
  for (int k = 0; k < 76; k += 4) {
    const int kk = k + 2 * half;
    v2f a, w;
    a.x = hp[kk];      // padded with zeros beyond NE
    a.y = hp[kk + 1];
    w.x = (o < 25 && kk     < NE) ? l1w[(i * 25 + o) * NE + kk]     : 0.f;
    w.y = (o < 25 && kk + 1 < NE) ? l1w[(i * 25 + o) * NE + kk + 1] : 0.f;
    c = wmma4(a, w, c);
  }
  const float bias = (o < 25) ? l1b[i * 25 + o] : 0.f;
#pragma unroll
  for (int r = 0; r < 8; ++r)
    if (o < 25)
      abuf[(size_t)(i * B + mt * 16 + r + 8 * half) * 32 + o] = c[r] + bias;
}

// ---- Stage 5: batchnorm over B, then l2 + relu, scatter into concat x -----
__global__ void bn_l2_kernel(const float* __restrict__ abuf,
                             const float* __restrict__ g1,
                             const float* __restrict__ be1,
                             const float* __restrict__ l2w,
                             const float* __restrict__ l2b,
                             float* __restrict__ xbuf) {
  const int i = blockIdx.x;
  const int tid = threadIdx.x;
  __shared__ float sa[B * 25];
  __shared__ float smean[25], sistd[25];
  for (int idx = tid; idx < B * 25; idx += blockDim.x)
    sa[idx] = abuf[(size_t)(i * B + idx / 25) * 32 + (idx % 25)];
  __syncthreads();
  if (tid < 25) {
    float s = 0.f, s2 = 0.f;
    for (int b = 0; b < B; ++b) { float v = sa[b * 25 + tid]; s += v; s2 += v * v; }
    const float mean = s / (float)B;
    const float var  = s2 / (float)B - mean * mean;   // population var
    smean[tid] = mean;
    sistd[tid] = rsqrtf(var + EPS);
  }
  __syncthreads();
  for (int idx = tid; idx < B * 25; idx += blockDim.x) {
    const int o = idx % 25;
    sa[idx] = (sa[idx] - smean[o]) * sistd[o] * g1[i * 25 + o] + be1[i * 25 + o];
  }
  __syncthreads();
  for (int idx = tid; idx < B * 25; idx += blockDim.x) {
    const int b = idx / 25, o2 = idx % 25;
    float acc = l2b[i * 25 + o2];
    for (int o = 0; o < 25; ++o)
      acc += l2w[(i * 25 + o2) * 25 + o] * sa[b * 25 + o];
    xbuf[(size_t)b * 200 + i * 25 + o2] = fmaxf(acc, 0.f);   // relu
  }
}

// ---- Stage 6: fc1, M=128 N=100 K=200 via WMMA -----------------------------
__global__ void fc1_kernel(const float* __restrict__ xbuf,
                           const float* __restrict__ w,
                           const float* __restrict__ bias,
                           float* __restrict__ ybuf) {
  const int wid = __builtin_amdgcn_readfirstlane(
      blockIdx.x * (blockDim.x >> 5) + (threadIdx.x >> 5));
  const int lane = threadIdx.x & 31;
  const int half = lane >> 4, l16 = lane & 15;
  const int nt = wid % 7, mt = wid / 7;
  if (mt >= 8) return;
  const int o = nt * 16 + l16;
  const float* xp = xbuf + (size_t)(mt * 16 + l16) * 200;
  v8f c = {};
  for (int k = 0; k < 200; k += 4) {
    const int kk = k + 2 * half;
    v2f a, wf;
    a.x = xp[kk]; a.y = xp[kk + 1];
    wf.x = (o < 100) ? w[o * 200 + kk]     : 0.f;
    wf.y = (o < 100) ? w[o * 200 + kk + 1] : 0.f;
    c = wmma4(a, wf, c);
  }
  const float bb = (o < 100) ? bias[o] : 0.f;
#pragma unroll
  for (int r = 0; r < 8; ++r)
    if (o < 100)
      ybuf[(size_t)(mt * 16 + r + 8 * half) * 112 + o] = c[r] + bb;
}

// ---- Stage 7: batchnorm over B (100 features), in place -------------------
__global__ void bn2_kernel(float* __restrict__ ybuf,
                           const float* __restrict__ g,
                           const float* __restrict__ be) {
  const int o = threadIdx.x;
  if (o >= 100) return;
  float s = 0.f, s2 = 0.f;
  for (int b = 0; b < B; ++b) { float v = ybuf[b * 112 + o]; s += v; s2 += v * v; }
  const float mean = s / (float)B;
  const float var  = s2 / (float)B - mean * mean;
  const float istd = rsqrtf(var + EPS);
  for (int b = 0; b < B; ++b)
    ybuf[b * 112 + o] = (ybuf[b * 112 + o] - mean) * istd * g[o] + be[o];
}

// ---- Stage 8: fc2, M=128 N=70 K=100 via WMMA, writes d_out ----------------
__global__ void fc2_kernel(const float* __restrict__ ybuf,
                           const float* __restrict__ w,
                           const float* __restrict__ bias,
                           float* __restrict__ out) {
  const int wid = __builtin_amdgcn_readfirstlane(
      blockIdx.x * (blockDim.x >> 5) + (threadIdx.x >> 5));
  const int lane = threadIdx.x & 31;
  const int half = lane >> 4, l16 = lane & 15;
  const int nt = wid % 5, mt = wid / 5;
  if (mt >= 8) return;
  const int o = nt * 16 + l16;
  const float* yp = ybuf + (size_t)(mt * 16 + l16) * 112;
  v8f c = {};
  for (int k = 0; k < 100; k += 4) {
    const int kk = k + 2 * half;
    v2f a, wf;
    a.x = yp[kk]; a.y = yp[kk + 1];     // kk+1 <= 99, pad cols never read
    wf.x = (o < 70) ? w[o * 100 + kk]     : 0.f;
    wf.y = (o < 70) ? w[o * 100 + kk + 1] : 0.f;
    c = wmma4(a, wf, c);
  }
  const float bb = (o < 70) ? bias[o] : 0.f;
#pragma unroll
  for (int r = 0; r < 8; ++r)
    if (o < 70)
      out[(size_t)(mt * 16 + r + 8 * half) * 70 + o] = c[r] + bb;
}

// ---------------------------------------------------------------------------
extern "C" void kernel_launch(void* const* d_in, const int* in_sizes, int n_in,
                              void* d_out, int out_size, void* d_ws, size_t ws_size,
                              hipStream_t stream) {
  (void)in_sizes; (void)n_in; (void)out_size; (void)ws_size;
  const float* points   = (const float*)d_in[0];
  const int*   lengths  = (const int*)  d_in[1];
  const float* centers  = (const float*)d_in[2];
  const float* sharp    = (const float*)d_in[3];
  const float* conv1_w  = (const float*)d_in[4];
  const float* conv2_w  = (const float*)d_in[5];
  const float* l1_w     = (const float*)d_in[6];
  const float* l1_b     = (const float*)d_in[7];
  const float* bn1_g    = (const float*)d_in[8];
  const float* bn1_b    = (const float*)d_in[9];
  const float* l2_w     = (const float*)d_in[10];
  const float* l2_b     = (const float*)d_in[11];
  const float* fc1_w    = (const float*)d_in[12];
  const float* fc1_b    = (const float*)d_in[13];
  const float* fc1_bn_g = (const float*)d_in[14];
  const float* fc1_bn_b = (const float*)d_in[15];
  const float* fc2_w    = (const float*)d_in[16];
  const float* fc2_b    = (const float*)d_in[17];

  // Workspace layout (floats): feat | sl | h | a | x | y   (~10 MB total)
  float*  ws   = (float*)d_ws;
  float4* feat = (float4*)ws;                          // V*B*P float4
  float*  sl   = ws   + (size_t)V * B * P * 4;         // V*3*B*NEP
  float*  hbuf = sl   + (size_t)V * 3 * B * NEP;       // V*B*NEP
  float*  abuf = hbuf + (size_t)V * B * NEP;           // V*B*32
  float*  xbuf = abuf + (size_t)V * B * 32;            // B*200
  float*  ybuf = xbuf + (size_t)B * 200;               // B*112

  feat_kernel<<<(V * B * P + 255) / 256, 256, 0, stream>>>(points, feat, V * B * P);
  resp_kernel<<<(V * 3 * B * 5) / 8, 256, 0, stream>>>(feat, lengths, centers, sharp, sl);
  conv_kernel<<<(V * B * NEP + 255) / 256, 256, 0, stream>>>(sl, conv1_w, conv2_w, hbuf);
  l1_kernel<<<16, 256, 0, stream>>>(hbuf, l1_w, l1_b, abuf);
  bn_l2_kernel<<<V, 256, 0, stream>>>(abuf, bn1_g, bn1_b, l2_w, l2_b, xbuf);
  fc1_kernel<<<7, 256, 0, stream>>>(xbuf, fc1_w, fc1_b, ybuf);
  bn2_kernel<<<1, 128, 0, stream>>>(ybuf, fc1_bn_g, fc1_bn_b);
  fc2_kernel<<<5, 256, 0, stream>>>(ybuf, fc2_w, fc2_b, (float*)d_out);
}